// FindPeaks_8435315769626
// MI455X (gfx1250) — compile-verified
//
#include <hip/hip_runtime.h>
#include <hip/hip_bf16.h>
#include <stdint.h>

// FindPeaks on [B,1,768,768] fp32 heatmaps.
// Memory-bound stencil: stage row tiles into LDS with the CDNA5 Tensor Data
// Mover (tensor_load_to_lds, wait via s_wait_tensorcnt), OOB rows zero-filled
// by the TDM (matches zero padding), then a 4-wide peak test per thread using
// ds_load_b128 row reads and one coalesced global_store_b128 per quad.

#define W_DIM 768
#define H_DIM 768
#define ROWS  8
#define BLOCK 256
#define QPR   (W_DIM / 4)          // quads per row = 192

typedef __attribute__((ext_vector_type(4))) unsigned int v4u;
typedef __attribute__((ext_vector_type(8))) unsigned int v8u;

// peak iff c > max(strict nbrs: NW,N,NE,W) and c >= max(nonstrict: E,SW,S,SE)
// (first-occurrence argmax tie-breaking; equivalent to per-neighbor compares
// for NaN-free inputs).
__device__ __forceinline__ float peak_val(float c, float nw, float n, float ne,
                                          float w, float e, float sw, float s,
                                          float se) {
    const float m1 = fmaxf(fmaxf(nw, n), fmaxf(ne, w));
    const float m2 = fmaxf(fmaxf(e, sw), fmaxf(s, se));
    return ((m1 < c) && (m2 <= c)) ? c : 0.0f;
}

__global__ __launch_bounds__(BLOCK)
void find_peaks_tdm_kernel(const float* __restrict__ in, float* __restrict__ out)
{
    __shared__ float tile[(ROWS + 2) * W_DIM];   // 30 KB: ROWS outputs + 2 halo rows

    const unsigned tid = threadIdx.x;
    const unsigned b   = blockIdx.y;
    const unsigned r0  = blockIdx.x * ROWS;      // first output row of this tile

    // LDS byte offset of tile[]: generic LDS pointers carry the wave-relative
    // LDS offset in their low 32 bits (ISA 10.2 aperture rules).
    const unsigned lds_off = (unsigned)(uintptr_t)(const void*)tile;

    // Tile spans global rows [r0-1, r0+ROWS]. Top edge: start at row 0, write
    // at LDS row 1, zero LDS row 0 manually. Bottom edge: tensor_dim1 = rows
    // remaining from tile start, so TDM OOB reads zero-fill -> zero padding.
    int g_start = (int)r0 - 1;
    unsigned lds_row0 = 0;
    if (g_start < 0) { g_start = 0; lds_row0 = 1; }
    const unsigned rows_req = (ROWS + 2u) - lds_row0;         // tile_dim1
    const unsigned tdim1    = (unsigned)(H_DIM - g_start);    // rows remaining

    const unsigned long long ga =
        (unsigned long long)(uintptr_t)in +
        ((unsigned long long)b * ((unsigned long long)H_DIM * W_DIM) +
         (unsigned long long)g_start * W_DIM) * 4ull;

    // ---- Tensor DMA descriptor (D#), groups 0 and 1 (2D tensor) ----
    v4u g0;
    g0[0] = 0x1u;                                    // count=1, gather off
    g0[1] = lds_off + lds_row0 * (W_DIM * 4u);       // lds_addr (bytes)
    g0[2] = (unsigned)ga;                            // global_addr[31:0]
    g0[3] = ((unsigned)(ga >> 32) & 0x01FFFFFFu)     // global_addr[56:32]
            | (2u << 30);                            // type = 2 ("image")

    v8u g1;
    g1[0] = 2u << 16;                                // data_size = 4B; no multicast/flags
    g1[1] = ((unsigned)W_DIM & 0xFFFFu) << 16;       // tensor_dim0[15:0]
    g1[2] = ((unsigned)W_DIM >> 16)                  // tensor_dim0[31:16]
            | ((tdim1 & 0xFFFFu) << 16);             // tensor_dim1[15:0]
    g1[3] = (tdim1 >> 16)                            // tensor_dim1[31:16]
            | ((unsigned)W_DIM << 16);               // tile_dim0 = 768
    g1[4] = rows_req;                                // tile_dim1; tile_dim2 = 0
    g1[5] = (unsigned)W_DIM;                         // tensor_dim0_stride[31:0]
    g1[6] = 0;                                       // stride hi / dim1_stride lo
    g1[7] = 0;                                       // dim1_stride hi (2D)

    // Zero top halo row for the first row-block (global row -1 == zero pad).
    if (r0 == 0) {
        for (unsigned i = tid; i < W_DIM; i += BLOCK) tile[i] = 0.0f;
    }

    // One wave issues the TDM transfer (EXEC ignored by tensor ops; the execz
    // branch keeps the other 7 waves from redundantly re-issuing it).
    if (tid < 32) {
        asm volatile("tensor_load_to_lds %0, %1" :: "s"(g0), "s"(g1) : "memory");
        __builtin_amdgcn_s_wait_tensorcnt(0);
    }
    __syncthreads();

    // ---- 4-wide peak test from LDS ----
    const unsigned base = b * (unsigned)(H_DIM * W_DIM) + r0 * W_DIM;
    for (unsigned qi = tid; qi < ROWS * QPR; qi += BLOCK) {
        const unsigned rr   = qi / QPR;
        const unsigned col4 = (qi - rr * QPR) * 4;

        const float* uprow = &tile[rr * W_DIM];
        const float* mdrow = uprow + W_DIM;
        const float* dnrow = mdrow + W_DIM;

        // Aligned 16B row reads (row stride 3072B, quad offset 16B-aligned).
        const float4 u = *(const float4*)&uprow[col4];
        const float4 m = *(const float4*)&mdrow[col4];
        const float4 d = *(const float4*)&dnrow[col4];

        // Halo scalars; x-boundary quads compare against the zero pad.
        const bool hasL = (col4 > 0);
        const bool hasR = (col4 < W_DIM - 4);
        const float uL = hasL ? uprow[col4 - 1] : 0.0f;
        const float mL = hasL ? mdrow[col4 - 1] : 0.0f;
        const float dL = hasL ? dnrow[col4 - 1] : 0.0f;
        const float uR = hasR ? uprow[col4 + 4] : 0.0f;
        const float mR = hasR ? mdrow[col4 + 4] : 0.0f;
        const float dR = hasR ? dnrow[col4 + 4] : 0.0f;

        float4 r;
        //                 c     NW    N     NE    W     E     SW    S     SE
        r.x = peak_val(m.x, uL,  u.x,  u.y,  mL,   m.y,  dL,   d.x,  d.y);
        r.y = peak_val(m.y, u.x, u.y,  u.z,  m.x,  m.z,  d.x,  d.y,  d.z);
        r.z = peak_val(m.z, u.y, u.z,  u.w,  m.y,  m.w,  d.y,  d.z,  d.w);
        r.w = peak_val(m.w, u.z, u.w,  uR,   m.z,  mR,   d.z,  d.w,  dR);

        *(float4*)&out[base + rr * W_DIM + col4] = r;   // coalesced b128 store
    }
}

extern "C" void kernel_launch(void* const* d_in, const int* in_sizes, int n_in,
                              void* d_out, int out_size, void* d_ws, size_t ws_size,
                              hipStream_t stream) {
    const float* in  = (const float*)d_in[0];
    float*       out = (float*)d_out;
    const int B = in_sizes[0] / (H_DIM * W_DIM);   // C == 1
    dim3 grid(H_DIM / ROWS, B);
    find_peaks_tdm_kernel<<<grid, dim3(BLOCK), 0, stream>>>(in, out);
    (void)n_in; (void)out_size; (void)d_ws; (void)ws_size;
}